// DetectionLoss_65171833750220
// MI455X (gfx1250) — compile-verified
//
#include <hip/hip_runtime.h>
#include <hip/hip_bf16.h>
#include <stdint.h>

// ---------------- constants ----------------
#define NB   16
#define NGT  32
#define NA   3
#define NC   80
#define THR  0.5f
#define NOOBJ_W 0.5f
#define EPSF 1e-7f

// workspace layout (4-byte units)
#define WS_OBJNEG  0   // float[3]  sum softplus(obj) over all cells
#define WS_OBJCORR 3   // float[3]  positive-cell obj corrections
#define WS_BOX     6   // float[3]  sum (1-ciou) over valid
#define WS_CLS     9   // float[3]  cls loss numerator
#define WS_NVALID  12  // int[3]
#define WS_NPOS    15  // int[3]
#define WS_CNT     18  // int[48]   per (scale,batch) record count
#define WS_REC     68  // int4[48*96] records {a, gy, gx, gt}
#define WS_INIT_WORDS 68

typedef __attribute__((ext_vector_type(2))) float v2f;
typedef __attribute__((ext_vector_type(8))) float v8f;
typedef __attribute__((ext_vector_type(4))) unsigned int v4u_t;
typedef __attribute__((ext_vector_type(8))) int v8i_t;
typedef __attribute__((ext_vector_type(4))) int v4i_t;

__device__ __forceinline__ float sp(float x) {           // bce(x,0) = softplus(x)
    return fmaxf(x, 0.0f) + log1pf(expf(-fabsf(x)));
}

// ---- exact 32-lane sum on the matrix pipe: two chained 16x16x4 f32 WMMAs ----
// WMMA1: A(vgpr0=p, vgpr1=0), B=ones -> D[m][n] = p[m] + p[m+16] = s_m
// adds : a0' = d[0..3] sums -> t0 (lanes0-15) / t2 (lanes16-31); a1' -> t1/t3
// WMMA2: A(a0', a1'), B=ones -> every D element = t0+t1+t2+t3 = full wave sum
__device__ __forceinline__ float wave_sum_wmma(float p) {
    v2f a;    a[0] = p;    a[1] = 0.0f;
    v2f ones; ones[0] = 1.0f; ones[1] = 1.0f;
    v8f c = {};
    v8f d1 = __builtin_amdgcn_wmma_f32_16x16x4_f32(false, a, false, ones,
                                                   (short)0, c, false, false);
    v2f a2;
    a2[0] = d1[0] + d1[1] + d1[2] + d1[3];
    a2[1] = d1[4] + d1[5] + d1[6] + d1[7];
    v8f d2 = __builtin_amdgcn_wmma_f32_16x16x4_f32(false, a2, false, ones,
                                                   (short)0, c, false, false);
    return d2[0];
}

// ---------------- K0: zero accumulators ----------------
__global__ void k_init(int* wsI) {
    int i = threadIdx.x;
    if (i < WS_INIT_WORDS) wsI[i] = 0;
}

// ---------------- K1: anchor assignment (one block per scale*batch) ----------------
__global__ __launch_bounds__(96) void k_assign(const float* __restrict__ gtb,
                                               const int*   __restrict__ gtl,
                                               const float* __restrict__ a0,
                                               const float* __restrict__ a1,
                                               const float* __restrict__ a2,
                                               int* wsI) {
    const int s = blockIdx.x >> 4;
    const int b = blockIdx.x & 15;
    const int H = (s == 0) ? 80 : ((s == 1) ? 40 : 20);
    const int W = H;
    const float stride = (s == 0) ? 8.0f : ((s == 1) ? 16.0f : 32.0f);
    const float* anc = (s == 0) ? a0 : ((s == 1) ? a1 : a2);

    __shared__ float sIou[96];
    __shared__ float sCand[96];
    __shared__ int   sKey[96];
    __shared__ int   sPos[96];
    __shared__ int   sCnt;

    const int tid = threadIdx.x;
    const int n = tid & 31;       // gt index
    const int a = tid >> 5;       // anchor index
    if (tid == 0) sCnt = 0;

    const float* g = gtb + (b * NGT + n) * 4;
    const float g0 = g[0], g1 = g[1], g2 = g[2], g3 = g[3];
    const float gcx = (g0 + g2) * 0.5f, gcy = (g1 + g3) * 0.5f;
    int gx = (int)(gcx / stride); gx = gx < 0 ? 0 : (gx > W - 1 ? W - 1 : gx);
    int gy = (int)(gcy / stride); gy = gy < 0 ? 0 : (gy > H - 1 ? H - 1 : gy);

    const float* A_ = anc + (((a * H) + gy) * W + gx) * 4;
    const float ax = A_[0], ay = A_[1], aw = A_[2], ah = A_[3];
    const float ax1 = ax - aw * 0.5f, ay1 = ay - ah * 0.5f;
    const float ax2 = ax + aw * 0.5f, ay2 = ay + ah * 0.5f;
    float iw = fminf(g2, ax2) - fmaxf(g0, ax1); iw = fmaxf(iw, 0.0f);
    float ih = fminf(g3, ay2) - fmaxf(g1, ay1); ih = fmaxf(ih, 0.0f);
    const float inter = iw * ih;
    const float iou = inter / ((g2 - g0) * (g3 - g1) + aw * ah - inter + EPSF);
    const int key = a * H * W + gy * W + gx;

    sIou[tid] = iou; sKey[tid] = key;
    __syncthreads();

    const float i0 = sIou[n], i1 = sIou[32 + n], i2 = sIou[64 + n];
    const bool has = (i0 > THR) || (i1 > THR) || (i2 > THR);
    int amax = 0; float mx = i0;
    if (i1 > mx) { mx = i1; amax = 1; }
    if (i2 > mx) { mx = i2; amax = 2; }
    const bool posF = has ? (iou > THR) : (a == amax);
    const float cand = posF ? iou : -1.0f;
    sCand[tid] = cand; sPos[tid] = posF ? 1 : 0;
    __syncthreads();

    float best = -1.0f;
    for (int j = 0; j < 96; ++j)
        if (sKey[j] == key) best = fmaxf(best, sCand[j]);
    const bool isb = posF && (cand >= best);
    int win = NGT;
    for (int j = 0; j < 96; ++j)
        if (sKey[j] == key && sPos[j] && sCand[j] >= best) {
            const int gj = j & 31;
            win = gj < win ? gj : win;
        }
    if (isb && n == win) {
        const int idx = atomicAdd(&sCnt, 1);
        int4 r; r.x = a; r.y = gy; r.z = gx; r.w = n;
        ((int4*)(wsI + WS_REC))[(s * NB + b) * 96 + idx] = r;
    }
    __syncthreads();
    if (tid == 0) wsI[WS_CNT + s * NB + b] = sCnt;
}

// ---------------- K2: obj-channel softplus sum via TDM strided gather ----------------
// 512-row tiles, 4 tiles per block (2048 rows) -> ~198 concurrent single-wave
// blocks across the three scales: enough parallelism to spread the ~52 MB of
// obj-channel cachelines over all shader engines while the TDM double-buffer
// hides DMA latency.
#define TDM_CHUNK 512
#define TDM_TILES 4

#if __has_builtin(__builtin_amdgcn_tensor_load_to_lds)
#define HAVE_TDM 1
#else
#define HAVE_TDM 0
#endif

#if HAVE_TDM
// DMA-gather `rows` objectness logits (stride 85 floats) into LDS at lds_byte_off.
__device__ __forceinline__ void tdm_gather_obj(const float* base, long long row0,
                                               int rows, unsigned lds_byte_off) {
    unsigned long long ga = (unsigned long long)(uintptr_t)base +
                            ((unsigned long long)row0 * 85ull + 4ull) * 4ull;
    v4u_t g0;
    g0[0] = 1u;                                   // count=1, user descriptor
    g0[1] = lds_byte_off;                         // lds_addr
    g0[2] = (unsigned)ga;                         // global_addr lo
    g0[3] = (unsigned)(ga >> 32) | (2u << 30);    // global_addr hi | type=2
    unsigned ur = (unsigned)rows;
    v8i_t g1;
    g1[0] = (int)(2u << 16);                      // data_size = 4B
    g1[1] = (int)(85u << 16);                     // tensor_dim0 = 85
    g1[2] = (int)((ur & 0xFFFFu) << 16);          // tensor_dim1 lo16
    g1[3] = (int)((ur >> 16) | (1u << 16));       // tensor_dim1 hi | tile_dim0=1
    g1[4] = (int)(ur & 0xFFFFu);                  // tile_dim1 = rows, tile_dim2=0
    g1[5] = 85;                                   // tensor_dim0_stride lo
    g1[6] = 0;
    g1[7] = 0;
    v4i_t gz = {0, 0, 0, 0};
#if defined(__clang_major__) && __clang_major__ >= 23
    v8i_t gz8 = {0, 0, 0, 0, 0, 0, 0, 0};
    __builtin_amdgcn_tensor_load_to_lds(g0, g1, gz, gz, gz8, 0);
#else
    __builtin_amdgcn_tensor_load_to_lds(g0, g1, gz, gz, 0);
#endif
}
#endif

__global__ __launch_bounds__(32) void k_objsum(const float* __restrict__ p,
                                               int cells, float* acc) {
    const int lane = threadIdx.x;
    float accl = 0.0f;
#if HAVE_TDM
    __shared__ __attribute__((aligned(16))) float tile[2][TDM_CHUNK];
    const long long row0 = (long long)blockIdx.x * (TDM_CHUNK * TDM_TILES);
    long long remain = (long long)cells - row0;
    if (remain < 0) remain = 0;
    const int total = remain < (TDM_CHUNK * TDM_TILES) ? (int)remain
                                                       : (TDM_CHUNK * TDM_TILES);
    const int nt = (total + TDM_CHUNK - 1) / TDM_CHUNK;
    const unsigned off0 = (unsigned)(uintptr_t)&tile[0][0];
    const unsigned off1 = (unsigned)(uintptr_t)&tile[1][0];

    if (nt > 0) tdm_gather_obj(p, row0, (total < TDM_CHUNK ? total : TDM_CHUNK), off0);
    if (nt > 1) {
        int r1 = total - TDM_CHUNK; if (r1 > TDM_CHUNK) r1 = TDM_CHUNK;
        tdm_gather_obj(p, row0 + TDM_CHUNK, r1, off1);
    }
    for (int t = 0; t < nt; ++t) {
        if (t + 1 < nt) {
#if __has_builtin(__builtin_amdgcn_s_wait_tensorcnt)
            __builtin_amdgcn_s_wait_tensorcnt(1);
#else
            asm volatile("s_wait_tensorcnt 0x1" ::: "memory");
#endif
        } else {
#if __has_builtin(__builtin_amdgcn_s_wait_tensorcnt)
            __builtin_amdgcn_s_wait_tensorcnt(0);
#else
            asm volatile("s_wait_tensorcnt 0x0" ::: "memory");
#endif
        }
        asm volatile("" ::: "memory");   // keep LDS reads below the wait
        int rows = total - t * TDM_CHUNK; if (rows > TDM_CHUNK) rows = TDM_CHUNK;
        const float* buf = &tile[t & 1][0];
        if (rows == TDM_CHUNK) {
#pragma unroll
            for (int k = 0; k < TDM_CHUNK / 128; ++k) {
                float4 v = *(const float4*)(buf + k * 128 + lane * 4);
                accl += sp(v.x) + sp(v.y) + sp(v.z) + sp(v.w);
            }
        } else {
            for (int i = lane; i < rows; i += 32) accl += sp(buf[i]);
        }
        asm volatile("" ::: "memory");   // reads retired before buffer reuse
        if (t + 2 < nt) {
            int rn = total - (t + 2) * TDM_CHUNK; if (rn > TDM_CHUNK) rn = TDM_CHUNK;
            tdm_gather_obj(p, row0 + (long long)(t + 2) * TDM_CHUNK, rn,
                           (t & 1) ? off1 : off0);
        }
    }
#else
    // Fallback: direct strided reads of the obj channel.
    const long long row0 = (long long)blockIdx.x * (TDM_CHUNK * TDM_TILES);
    for (long long r = row0 + lane; r < row0 + TDM_CHUNK * TDM_TILES && r < cells; r += 32)
        accl += sp(__builtin_nontemporal_load(p + r * 85 + 4));
#endif
    // all 32 lanes active & converged here: EXEC all-1s as WMMA requires
    const float wsum = wave_sum_wmma(accl);
    if (lane == 0) atomicAdd(acc, wsum);
}

// ---------------- K3: sparse positive cells: obj correction, box CIoU, cls ----------------
__global__ __launch_bounds__(128) void k_pos(const float* __restrict__ p0,
                                             const float* __restrict__ p1,
                                             const float* __restrict__ p2,
                                             const float* __restrict__ a0,
                                             const float* __restrict__ a1,
                                             const float* __restrict__ a2,
                                             const float* __restrict__ gtb,
                                             const int*   __restrict__ gtl,
                                             int* wsI, float* wsF) {
    const int s = blockIdx.x >> 4;
    const int b = blockIdx.x & 15;
    const int H = (s == 0) ? 80 : ((s == 1) ? 40 : 20);
    const int W = H;
    const float stride = (s == 0) ? 8.0f : ((s == 1) ? 16.0f : 32.0f);
    const float* p   = (s == 0) ? p0 : ((s == 1) ? p1 : p2);
    const float* anc = (s == 0) ? a0 : ((s == 1) ? a1 : a2);

    const int cnt = wsI[WS_CNT + s * NB + b];
    const int r = threadIdx.x;
    if (r >= cnt) return;

    const int4 rec = ((const int4*)(wsI + WS_REC))[(s * NB + b) * 96 + r];
    const int a = rec.x, gy = rec.y, gx = rec.z, n = rec.w;
    const long long cell = ((((long long)b * NA + a) * H + gy) * W + gx);
    const float* pc = p + cell * 85;

    // --- obj correction: bce(x,1)*1 - bce(x,0)*NOOBJ_W on top of global neg sum ---
    const float xo = pc[4];
    atomicAdd(wsF + WS_OBJCORR + s, sp(-xo) - NOOBJ_W * sp(xo));
    atomicAdd(wsI + WS_NPOS + s, 1);

    // --- box: decode + CIoU vs winner gt ---
    const float* A_ = anc + ((a * H + gy) * W + gx) * 4;
    const float acx = A_[0], acy = A_[1], aw = A_[2], ah = A_[3];
    float s0 = 1.0f / (1.0f + expf(-pc[0]));
    float s1 = 1.0f / (1.0f + expf(-pc[1]));
    float s2 = 1.0f / (1.0f + expf(-pc[2]));
    float s3 = 1.0f / (1.0f + expf(-pc[3]));
    const float cx = acx + (s0 * 2.0f - 0.5f) * stride;
    const float cy = acy + (s1 * 2.0f - 0.5f) * stride;
    const float bw = aw * (s2 * 2.0f) * (s2 * 2.0f);
    const float bh = ah * (s3 * 2.0f) * (s3 * 2.0f);
    const float b1x1 = cx - bw * 0.5f, b1y1 = cy - bh * 0.5f;
    const float b1x2 = cx + bw * 0.5f, b1y2 = cy + bh * 0.5f;

    const float* g = gtb + (b * NGT + n) * 4;
    const float b2x1 = g[0], b2y1 = g[1], b2x2 = g[2], b2y2 = g[3];

    const float w1 = b1x2 - b1x1, h1 = b1y2 - b1y1;
    const float w2 = b2x2 - b2x1, h2 = b2y2 - b2y1;
    float iw = fminf(b1x2, b2x2) - fmaxf(b1x1, b2x1); iw = fmaxf(iw, 0.0f);
    float ih = fminf(b1y2, b2y2) - fmaxf(b1y1, b2y1); ih = fmaxf(ih, 0.0f);
    const float inter = iw * ih;
    const float uni = w1 * h1 + w2 * h2 - inter + EPSF;
    const float iou = inter / uni;
    const float cw = fmaxf(b1x2, b2x2) - fminf(b1x1, b2x1);
    const float ch = fmaxf(b1y2, b2y2) - fminf(b1y1, b2y1);
    const float c2 = cw * cw + ch * ch + EPSF;
    const float dx = b2x1 + b2x2 - b1x1 - b1x2;
    const float dy = b2y1 + b2y2 - b1y1 - b1y2;
    const float rho2 = (dx * dx + dy * dy) * 0.25f;
    const float pi2 = 9.8696044010893586f;           // pi^2
    const float dat = atanf(w2 / (h2 + EPSF)) - atanf(w1 / (h1 + EPSF));
    const float v = (4.0f / pi2) * dat * dat;
    const float alpha = v / (v - iou + (1.0f + EPSF));
    const float ciou = iou - rho2 / c2 - alpha * v;
    if (isfinite(ciou)) {
        atomicAdd(wsF + WS_BOX + s, 1.0f - ciou);
        atomicAdd(wsI + WS_NVALID + s, 1);
    }

    // --- cls: sum_c softplus(x_c) - x_label ---
    float cacc = 0.0f;
    for (int c = 0; c < NC; ++c) cacc += sp(pc[5 + c]);
    cacc -= pc[5 + gtl[b * NGT + n]];
    atomicAdd(wsF + WS_CLS + s, cacc);
}

// ---------------- K4: finalize scalar ----------------
__global__ void k_final(const int* wsI, const float* wsF, float* out) {
    if (threadIdx.x != 0 || blockIdx.x != 0) return;
    float bl = 0.0f, ol = 0.0f, cl = 0.0f;
    int tv = 0;
    for (int s = 0; s < 3; ++s) {
        const int H = (s == 0) ? 80 : ((s == 1) ? 40 : 20);
        const float ncells = (float)(NB * NA * H * H);
        const int npos = wsI[WS_NPOS + s];
        const int nval = wsI[WS_NVALID + s];
        const float wgt_sum = NOOBJ_W * ncells + (1.0f - NOOBJ_W) * (float)npos;
        const float obj = (NOOBJ_W * wsF[WS_OBJNEG + s] + wsF[WS_OBJCORR + s]) /
                          (wgt_sum + 1e-6f);
        const float box = wsF[WS_BOX + s] / (float)(nval > 1 ? nval : 1);
        const int cden = npos * NC;
        const float cls = wsF[WS_CLS + s] / (float)(cden > 1 ? cden : 1);
        bl += box; ol += obj; cl += cls; tv += nval;
    }
    if (tv > 0) { bl /= 3.0f; cl /= 3.0f; }
    ol /= 3.0f;
    const float total = 0.15f * bl + 5.0f * ol + 0.5f * cl;
    out[0] = isfinite(total) ? total : 0.0f;
}

// ---------------- launch ----------------
extern "C" void kernel_launch(void* const* d_in, const int* in_sizes, int n_in,
                              void* d_out, int out_size, void* d_ws, size_t ws_size,
                              hipStream_t stream) {
    const float* p0  = (const float*)d_in[0];
    const float* p1  = (const float*)d_in[1];
    const float* p2  = (const float*)d_in[2];
    const float* a0  = (const float*)d_in[3];
    const float* a1  = (const float*)d_in[4];
    const float* a2  = (const float*)d_in[5];
    const float* gtb = (const float*)d_in[6];
    const int*   gtl = (const int*)d_in[7];
    float* out = (float*)d_out;
    float* wsF = (float*)d_ws;
    int*   wsI = (int*)d_ws;
    (void)in_sizes; (void)n_in; (void)out_size; (void)ws_size;

    k_init<<<1, 96, 0, stream>>>(wsI);
    k_assign<<<48, 96, 0, stream>>>(gtb, gtl, a0, a1, a2, wsI);

    const int cells0 = NB * NA * 80 * 80;   // 307200
    const int cells1 = NB * NA * 40 * 40;   // 76800
    const int cells2 = NB * NA * 20 * 20;   // 19200
    const int rpb = TDM_CHUNK * TDM_TILES;  // rows per block
    k_objsum<<<(cells0 + rpb - 1) / rpb, 32, 0, stream>>>(p0, cells0, wsF + WS_OBJNEG + 0);
    k_objsum<<<(cells1 + rpb - 1) / rpb, 32, 0, stream>>>(p1, cells1, wsF + WS_OBJNEG + 1);
    k_objsum<<<(cells2 + rpb - 1) / rpb, 32, 0, stream>>>(p2, cells2, wsF + WS_OBJNEG + 2);

    k_pos<<<48, 128, 0, stream>>>(p0, p1, p2, a0, a1, a2, gtb, gtl, wsI, wsF);
    k_final<<<1, 32, 0, stream>>>(wsI, wsF, out);
}